// HigherPointV2_6846177869889
// MI455X (gfx1250) — compile-verified
//
#include <hip/hip_runtime.h>
#include <hip/hip_bf16.h>

// CDNA5 / gfx1250: wave32, WMMA f32 16x16x4.
// Memory-bound problem (~400MB min traffic @ 23.3TB/s => ~17us floor, only ~2.1
// GFLOP of GEMM), so: keep f32 precision, read fw exactly once (dual N-tile
// accumulators), non-temporal policy on the streaming tensors (fw read-once,
// out write-once) so L2 keeps the reused pair_features.

typedef __attribute__((ext_vector_type(2))) float v2f;
typedef __attribute__((ext_vector_type(8))) float v8f;

#define P_ 128
#define C_ 16
#define S_ 32
#define X_ 16
#define T_ 16

__global__ __launch_bounds__(256, 2)
void higherpoint_v3_kernel(const float* __restrict__ fw,    // [P,C,S,S,X]
                           const float* __restrict__ pf,    // [P,T,S,X]
                           const float* __restrict__ w0,    // [T,C]
                           const float* __restrict__ w1,    // [T,C]
                           const float* __restrict__ w2,    // [T,C]
                           const float* __restrict__ bias,  // [C]
                           float* __restrict__ out)         // [P,C,S,S,S]
{
    const int pz = blockIdx.x;        // p*16 + z
    const int p  = pz >> 4;
    const int z  = pz & 15;

    // Per-(p,z) projected pair features, LDS resident (3 * 2KB).
    __shared__ __align__(16) float sA[S_ * X_];   // A_z[i,x]
    __shared__ __align__(16) float sB[S_ * X_];   // B_z[j,x]
    __shared__ __align__(16) float sH[S_ * X_];   // H_z[y,x]

    const int tid = threadIdx.x;

    // ---- Stage 1: A/B/H = pf[p] contracted with w0/w1/w2 over T (VALU, tiny) ----
    {
        const float* pfp = pf + (size_t)p * (T_ * S_ * X_);
        for (int e = tid; e < S_ * X_; e += 256) {
            const int i = e >> 4;      // spatial index
            const int x = e & 15;      // contracted index
            float sa = 0.f, sb = 0.f, sh = 0.f;
            const float* src = pfp + i * X_ + x;
#pragma unroll
            for (int t = 0; t < T_; ++t) {
                const float v = src[t * (S_ * X_)];   // regular load: pf reused across z, keep in L2
                sa = fmaf(v, w0[t * C_ + z], sa);
                sb = fmaf(v, w1[t * C_ + z], sb);
                sh = fmaf(v, w2[t * C_ + z], sh);
            }
            sA[e] = sa; sB[e] = sb; sH[e] = sh;
        }
    }
    __syncthreads();

    // ---- Stage 2: out[ij, y] = (fw*A*B)[ij, x] @ H[y, x]^T via WMMA f32 16x16x4 ----
    const int lane = tid & 31;
    const int wave = tid >> 5;        // 0..7
    const int m    = lane & 15;       // M row (A) / N col (B,D) within tile
    const int hi   = lane >> 4;       // half-wave: K pairs {0,1} vs {2,3}
    const float bz = bias[z];

    const float* fwbase = fw  + (size_t)pz * (S_ * S_ * X_);
    float*       obase  = out + (size_t)pz * (S_ * S_ * S_);

    // B-fragments for both N-tiles are loop-invariant: hoist into VGPRs once.
    // B 4x16 layout mirrors A striping: lane n holds K = ks*4 + hi*2 + {0,1}.
    v2f bv0[4], bv1[4];
#pragma unroll
    for (int ks = 0; ks < 4; ++ks) {
        const int k0 = ks * 4 + hi * 2;
        bv0[ks] = *(const v2f*)(sH + m * X_ + k0);          // y = m        (N-tile 0)
        bv1[ks] = *(const v2f*)(sH + (16 + m) * X_ + k0);   // y = 16 + m   (N-tile 1)
    }

    // 64 M-tiles (16 flat-ij rows each => fixed i, j in [j0,j0+16)); both N-tiles per pass
    // so each fw element is loaded exactly once.
    for (int mt = wave; mt < 64; mt += 8) {
        const int i  = mt >> 1;
        const int j0 = (mt & 1) << 4;
        const int j  = j0 + m;
        const float* fwrow = fwbase + (i * S_ + j) * X_;
        const float* arow  = sA + i * X_;
        const float* brow  = sB + j * X_;

        // Prefetch next M-tile's fw row (uniform predicate: EXEC stays all-1s at WMMAs).
        if (mt + 8 < 64) {
            const int mt2 = mt + 8;
            const int i2  = mt2 >> 1;
            const int j2  = ((mt2 & 1) << 4) + m;
            __builtin_prefetch(fwbase + (i2 * S_ + j2) * X_, 0, 0);
        }

        v8f acc0 = {};
        v8f acc1 = {};
#pragma unroll
        for (int ks = 0; ks < 4; ++ks) {
            const int k0 = ks * 4 + hi * 2;   // A 16x4 layout: lanes0-15 K={0,1}, lanes16-31 K={2,3}
            const v2f g  = __builtin_nontemporal_load((const v2f*)(fwrow + k0));  // fw: read-once stream
            const v2f a2 = *(const v2f*)(arow + k0);
            const v2f b2 = *(const v2f*)(brow + k0);
            v2f av;
            av.x = g.x * a2.x * b2.x;         // fused G = fw * A * B
            av.y = g.y * a2.y * b2.y;
            acc0 = __builtin_amdgcn_wmma_f32_16x16x4_f32(
                false, av, false, bv0[ks], (short)0, acc0, false, false);
            acc1 = __builtin_amdgcn_wmma_f32_16x16x4_f32(
                false, av, false, bv1[ks], (short)0, acc1, false, false);
        }

        // D layout: VGPR v -> M = v + 8*hi, N = lane&15. Bias + ReLU, non-temporal stores
        // (out is write-once; keep it out of L2).
#pragma unroll
        for (int v = 0; v < 8; ++v) {
            const int M = v + (hi << 3);      // row within tile = j offset
            float* orow = obase + (i * S_ + (j0 + M)) * S_;
            float v0 = acc0[v] + bz; v0 = v0 > 0.f ? v0 : 0.f;
            float v1 = acc1[v] + bz; v1 = v1 > 0.f ? v1 : 0.f;
            __builtin_nontemporal_store(v0, orow + m);        // y = m
            __builtin_nontemporal_store(v1, orow + 16 + m);   // y = 16 + m
        }
    }
}

extern "C" void kernel_launch(void* const* d_in, const int* in_sizes, int n_in,
                              void* d_out, int out_size, void* d_ws, size_t ws_size,
                              hipStream_t stream) {
    const float* fw   = (const float*)d_in[0];  // features_weighted [P,C,S,S,X]
    const float* pf   = (const float*)d_in[1];  // pair_features    [P,T,S,X]
    const float* w0   = (const float*)d_in[2];  // [T,C]
    const float* w1   = (const float*)d_in[3];  // [T,C]
    const float* w2   = (const float*)d_in[4];  // [T,C]
    const float* bias = (const float*)d_in[5];  // [C]
    float* out = (float*)d_out;                 // [P,C,S,S,S]

    (void)in_sizes; (void)n_in; (void)out_size; (void)d_ws; (void)ws_size;

    dim3 grid(P_ * C_);   // one workgroup per (p, z)
    dim3 block(256);      // 8 waves of 32
    hipLaunchKernelGGL(higherpoint_v3_kernel, grid, block, 0, stream,
                       fw, pf, w0, w1, w2, bias, out);
}